// PRGNN_21268678050012
// MI455X (gfx1250) — compile-verified
//
#include <hip/hip_runtime.h>

// Problem constants (from reference): B=16, N=96, S=4, FIN=8, H=32, P=256
#define B_   16
#define N_   96
#define S_   4
#define FIN_ 8
#define H_   32
#define P_   256

typedef __attribute__((ext_vector_type(2))) float v2f;
typedef __attribute__((ext_vector_type(8))) float v8f;

__device__ __forceinline__ v8f wmma_f32_16x16x4(v2f a, v2f b, v8f c) {
    // 8-arg form: (neg_a, A, neg_b, B, c_mod, C, reuse_a, reuse_b)
    return __builtin_amdgcn_wmma_f32_16x16x4_f32(false, a, false, b, (short)0, c, false, false);
}

// One block per batch element b. 256 threads = 8 wave32.
// Phase 1: load x -> h0, mask into LDS
// Per layer: prep Mt[i][c][s] (=sum_f W[s,c*fin+f]*h[i,f]) and Bct[c][i] (=sum_f b[c*fin+f]*h[i,f])
//            then WMMA message pass (edge term + bias-proj term + root term) + bias/mask/relu
// Finally: pool over nodes, score[b] = relu(pooled @ Wd + bd) into workspace.
__global__ __launch_bounds__(256, 1)
void gnn_fused(const float* __restrict__ x, const float* __restrict__ adj,
               const float* __restrict__ e,
               const float* __restrict__ W1, const float* __restrict__ b1,
               const float* __restrict__ root1, const float* __restrict__ bias1,
               const float* __restrict__ W2, const float* __restrict__ b2,
               const float* __restrict__ root2, const float* __restrict__ bias2,
               const float* __restrict__ Wd, const float* __restrict__ bd,
               float* __restrict__ score)
{
    __shared__ float sh_hA[N_ * H_];        // 12 KB  (layer1 input / layer2 output)
    __shared__ float sh_hB[N_ * H_];        // 12 KB  (layer1 output / layer2 input)
    __shared__ float sh_mask[N_];
    __shared__ float sh_Mt[N_ * H_ * S_];   // 48 KB  Mt[(i*32+c)*4 + s]
    __shared__ float sh_Bct[H_ * N_];       // 12 KB  Bct[c*96 + i]
    __shared__ float sh_roott[H_ * H_];     //  4 KB  roott[c*32 + f]
    __shared__ float sh_bias[H_];
    __shared__ float sh_red[H_];

    const int tid  = (int)threadIdx.x;
    const int b    = (int)blockIdx.x;
    const int wave = tid >> 5;       // wave32
    const int lane = tid & 31;
    const int half = lane >> 4;      // K-half select for A/B fragments
    const int l16  = lane & 15;
    const int k0   = half * 2;       // first of the two K slots this lane holds

    // ---- load node features (first 8 cols of hA) and mask ----
    for (int idx = tid; idx < N_ * (FIN_ + 1); idx += 256) {
        int i = idx / (FIN_ + 1);
        int f = idx - i * (FIN_ + 1);
        float v = x[(b * N_ + i) * (FIN_ + 1) + f];
        if (f < FIN_) sh_hA[i * H_ + f] = v;
        else          sh_mask[i] = v;
    }
    __syncthreads();

    float* hin  = sh_hA;
    float* hout = sh_hB;

    for (int layer = 0; layer < 2; ++layer) {
        const int fin = (layer == 0) ? FIN_ : H_;
        const float* W  = (layer == 0) ? W1    : W2;     // (S, fin*H), col index = c*fin+f
        const float* bv = (layer == 0) ? b1    : b2;     // (fin*H,)
        const float* rt = (layer == 0) ? root1 : root2;  // (fin, H)
        const float* bs = (layer == 0) ? bias1 : bias2;  // (H,)

        // ---- prep: Mt, Bct, roott, bias ----
        for (int idx = tid; idx < N_ * H_; idx += 256) {
            int i = idx >> 5;
            int c = idx & 31;
            float m0 = 0.f, m1 = 0.f, m2 = 0.f, m3 = 0.f, bc = 0.f;
            for (int f = 0; f < fin; ++f) {
                float hv = hin[i * H_ + f];
                int col = c * fin + f;
                m0 = fmaf(W[0 * (H_ * fin) + col], hv, m0);
                m1 = fmaf(W[1 * (H_ * fin) + col], hv, m1);
                m2 = fmaf(W[2 * (H_ * fin) + col], hv, m2);
                m3 = fmaf(W[3 * (H_ * fin) + col], hv, m3);
                bc = fmaf(bv[col], hv, bc);
            }
            float* mp = &sh_Mt[(i * H_ + c) * S_];
            mp[0] = m0; mp[1] = m1; mp[2] = m2; mp[3] = m3;
            sh_Bct[c * N_ + i] = bc;
        }
        for (int idx = tid; idx < H_ * fin; idx += 256) {
            int f = idx >> 5;   // row of root (fin x H)
            int c = idx & 31;
            sh_roott[c * H_ + f] = rt[f * H_ + c];
        }
        if (tid < H_) sh_bias[tid] = bs[tid];
        __syncthreads();

        // ---- message pass: 12 WMMA jobs = 6 node-tiles x 2 channel-tiles ----
        for (int job = wave; job < 12; job += 8) {   // wave-uniform loop: EXEC all-1s
            int t  = job >> 1;            // node tile
            int ct = job & 1;             // channel tile
            int n  = t * 16 + l16;        // A-side row (node) for this lane
            int c  = ct * 16 + l16;       // B-side col (channel) for this lane

            v8f acc = {0.f, 0.f, 0.f, 0.f, 0.f, 0.f, 0.f, 0.f};

            const float* arow = adj + (b * N_ + n) * N_;
            const float* erow = e + (size_t)(b * N_ + n) * N_ * S_;

            // (1) edge-conditioned term: one K=4 WMMA per neighbor i (K dim = s)
            for (int i = 0; i < N_; ++i) {
                float av = arow[i];
                v2f af, bf;
                af.x = av * erow[i * S_ + k0 + 0];
                af.y = av * erow[i * S_ + k0 + 1];
                const float* mp = &sh_Mt[(i * H_ + c) * S_ + k0];
                bf.x = mp[0];
                bf.y = mp[1];
                acc = wmma_f32_16x16x4(af, bf, acc);
            }
            // (2) bias-projection term: a (N x N) @ Bc (N x H), K=4 chunks over i
            for (int i0 = 0; i0 < N_; i0 += 4) {
                v2f af, bf;
                af.x = arow[i0 + k0 + 0];
                af.y = arow[i0 + k0 + 1];
                const float* bp = &sh_Bct[c * N_ + i0 + k0];
                bf.x = bp[0];
                bf.y = bp[1];
                acc = wmma_f32_16x16x4(af, bf, acc);
            }
            // (3) root term: h (N x fin) @ root (fin x H), K=4 chunks over f
            for (int f0 = 0; f0 < fin; f0 += 4) {
                v2f af, bf;
                af.x = hin[n * H_ + f0 + k0 + 0];
                af.y = hin[n * H_ + f0 + k0 + 1];
                const float* rp = &sh_roott[c * H_ + f0 + k0];
                bf.x = rp[0];
                bf.y = rp[1];
                acc = wmma_f32_16x16x4(af, bf, acc);
            }

            // epilogue: C/D layout -> VGPR v: lanes<16 hold M=v, lanes>=16 hold M=8+v
            for (int v = 0; v < 8; ++v) {
                int node = t * 16 + half * 8 + v;
                int ch   = ct * 16 + l16;
                float val = (acc[v] + sh_bias[ch]) * sh_mask[node];
                hout[node * H_ + ch] = fmaxf(val, 0.f);
            }
        }
        __syncthreads();
        float* tmp = hin; hin = hout; hout = tmp;   // ping-pong
    }

    // ---- pool + score (hin == final h2) ----
    if (tid < H_) {
        float s = 0.f;
        for (int i = 0; i < N_; ++i) s += hin[i * H_ + tid];
        sh_red[tid] = s * Wd[tid];
    }
    __syncthreads();
    if (tid == 0) {
        float s = bd[0];
        for (int c = 0; c < H_; ++c) s += sh_red[c];
        score[b] = fmaxf(s, 0.f);
    }
}

__global__ void pref_diff(const float* __restrict__ score,
                          const int* __restrict__ pa, const int* __restrict__ pb,
                          float* __restrict__ out)
{
    int p = (int)(blockIdx.x * blockDim.x + threadIdx.x);
    if (p < P_) out[p] = score[pa[p]] - score[pb[p]];
}

extern "C" void kernel_launch(void* const* d_in, const int* in_sizes, int n_in,
                              void* d_out, int out_size, void* d_ws, size_t ws_size,
                              hipStream_t stream)
{
    (void)in_sizes; (void)n_in; (void)out_size; (void)ws_size;
    const float* x     = (const float*)d_in[0];
    const float* a     = (const float*)d_in[1];
    const float* e     = (const float*)d_in[2];
    const int*   pa    = (const int*)d_in[3];
    const int*   pb    = (const int*)d_in[4];
    const float* W1    = (const float*)d_in[5];
    const float* b1    = (const float*)d_in[6];
    const float* root1 = (const float*)d_in[7];
    const float* bias1 = (const float*)d_in[8];
    const float* W2    = (const float*)d_in[9];
    const float* b2    = (const float*)d_in[10];
    const float* root2 = (const float*)d_in[11];
    const float* bias2 = (const float*)d_in[12];
    const float* Wd    = (const float*)d_in[13];
    const float* bd    = (const float*)d_in[14];

    float* score = (float*)d_ws;  // 16 floats of scratch

    gnn_fused<<<B_, 256, 0, stream>>>(x, a, e, W1, b1, root1, bias1,
                                      W2, b2, root2, bias2, Wd, bd, score);
    pref_diff<<<1, 256, 0, stream>>>(score, pa, pb, (float*)d_out);
}